// FrameLabeller_8237747273827
// MI455X (gfx1250) — compile-verified
//
#include <hip/hip_runtime.h>
#include <math.h>

#define DIMS 100
#define LDKQV 300
#define NFR 1200
#define NRL 32
#define NEG_INF_ENC 0x007FFFFFu   // encF(-inf)

typedef __attribute__((ext_vector_type(2))) float v2f;
typedef __attribute__((ext_vector_type(8))) float v8f;

// ---- order-preserving float <-> uint encoding (for atomic max on floats) ----
__device__ __forceinline__ unsigned encF(float f){
  unsigned b = __float_as_uint(f);
  return (b & 0x80000000u) ? ~b : (b | 0x80000000u);
}
__device__ __forceinline__ float decF(unsigned u){
  unsigned b = (u & 0x80000000u) ? (u & 0x7FFFFFFFu) : ~u;
  return __uint_as_float(b);
}
__device__ __forceinline__ float geluf(float x){   // exact gelu (approximate=False)
  return 0.5f * x * (1.0f + erff(x * 0.7071067811865476f));
}

__global__ void fill_u32(unsigned* __restrict__ p, unsigned v, int n){
  int i = blockIdx.x * blockDim.x + threadIdx.x;
  if (i < n) p[i] = v;
}

__global__ void gelu_inplace(float* __restrict__ x, long n){
  long i = (long)blockIdx.x * blockDim.x + threadIdx.x;
  if (i < n) x[i] = geluf(x[i]);
}

// =====================================================================
// K=100 fp32 WMMA GEMM:  C[M,N] = gather(A) @ B + bias
// Wave computes a (16*MT) x (16*NT) tile via V_WMMA_F32_16X16X4_F32.
// LDB is a template constant and the k-loop is fully unrolled, so every
// A/B fragment load is {per-lane base + immediate offset}: the inner loop
// is pure vmem + wmma with zero address arithmetic and no spills
// (MT=2,NT=4: 64 acc VGPRs + 12 base VGPRs).
// OOB rows/cols are index-clamped: they only affect never-stored outputs,
// so there are no conditionals in the hot path (EXEC all-ones throughout).
// =====================================================================
template<int MT, int NT, int LDB>
__global__ void gemm_k100_wmma(const float* __restrict__ A, int lda,
                               const int* __restrict__ gidx,
                               const float* __restrict__ B,
                               const float* __restrict__ bias,
                               float* __restrict__ C, int ldc,
                               int M, int N){
  const int lane = threadIdx.x & 31;
  const int wave = threadIdx.x >> 5;
  const int sm   = blockIdx.x * (blockDim.x >> 5) + wave;   // m-supertile
  const int row0 = sm * (16 * MT);
  if (row0 >= M) return;                    // wave-uniform exit
  const int col0 = blockIdx.y * (16 * NT);
  const int half = lane >> 4;               // 0: lanes 0-15, 1: lanes 16-31
  const int l15  = lane & 15;

  // Per-lane base pointers; all k-step offsets become compile-time immediates.
  const float* Abase[MT];
  #pragma unroll
  for (int mt = 0; mt < MT; ++mt){
    int arow = row0 + mt * 16 + l15;
    if (arow >= M) arow = M - 1;            // clamp: OOB rows never stored
    const int asrc = gidx ? gidx[arow] : arow;
    Abase[mt] = A + (long)asrc * lda + half * 2;
  }
  const float* Bbase[NT];
  #pragma unroll
  for (int t = 0; t < NT; ++t){
    int c = col0 + t * 16 + l15;
    if (c >= N) c = N - 1;                  // clamp: OOB cols never stored
    Bbase[t] = B + (long)half * 2 * LDB + c;
  }

  v8f acc[MT][NT];
  v8f zeroAcc = {};
  #pragma unroll
  for (int mt = 0; mt < MT; ++mt)
    #pragma unroll
    for (int t = 0; t < NT; ++t) acc[mt][t] = zeroAcc;

  #pragma unroll
  for (int kk = 0; kk < DIMS; kk += 4){
    v2f af[MT];
    #pragma unroll
    for (int mt = 0; mt < MT; ++mt)
      af[mt] = *(const v2f*)(Abase[mt] + kk);          // imm offset kk*4
    v2f bfr[NT];
    #pragma unroll
    for (int t = 0; t < NT; ++t){
      bfr[t].x = Bbase[t][kk * LDB];                   // imm offset
      bfr[t].y = Bbase[t][kk * LDB + LDB];             // imm offset
    }
    #pragma unroll
    for (int mt = 0; mt < MT; ++mt)
      #pragma unroll
      for (int t = 0; t < NT; ++t)
        acc[mt][t] = __builtin_amdgcn_wmma_f32_16x16x4_f32(
                       false, af[mt], false, bfr[t], (short)0, acc[mt][t], false, false);
  }

  #pragma unroll
  for (int mt = 0; mt < MT; ++mt){
    const int rbase = row0 + mt * 16 + half * 8;   // C: VGPR j -> rows j (lo) / j+8 (hi)
    #pragma unroll
    for (int t = 0; t < NT; ++t){
      const int col = col0 + t * 16 + l15;
      if (col >= N) continue;
      const float bv = bias ? bias[col] : 0.0f;
      #pragma unroll
      for (int j = 0; j < 8; ++j){
        const int r = rbase + j;
        if (r < M) C[(long)r * ldc + col] = acc[mt][t][j] + bv;
      }
    }
  }
}

// =====================================================================
// HGT attention logits: a0 = <q_n[dst], k0[src]>*p0/10 ; a2 = <q_n[dst], k2[e]>*p2/10
// One wave per edge; also feeds segment max (by dst) via encoded atomicMax.
// =====================================================================
__global__ void attn_logits(const float* __restrict__ kqv_n,
                            const float* __restrict__ k0, const float* __restrict__ k2,
                            const int* __restrict__ esrc, const int* __restrict__ edst,
                            const float* __restrict__ p_rel,
                            float* __restrict__ a0, float* __restrict__ a2,
                            unsigned* __restrict__ segmax, int ne){
  int e = blockIdx.x * (blockDim.x >> 5) + (threadIdx.x >> 5);
  if (e >= ne) return;
  int lane = threadIdx.x & 31;
  int s = esrc[e], d = edst[e];
  const float* q   = kqv_n + (long)d * LDKQV + DIMS;   // q_n slice
  const float* pk0 = k0 + (long)s * DIMS;
  const float* pk2 = k2 + (long)e * DIMS;
  float s0 = 0.0f, s2 = 0.0f;
  for (int i = lane; i < DIMS; i += 32){
    float qv = q[i];
    s0 += qv * pk0[i];
    s2 += qv * pk2[i];
  }
  for (int o = 16; o; o >>= 1){ s0 += __shfl_xor(s0, o, 32); s2 += __shfl_xor(s2, o, 32); }
  if (lane == 0){
    const float sc = 0.1f;                    // 1/sqrt(100)
    float l0 = s0 * p_rel[0] * sc;
    float l2 = s2 * p_rel[2] * sc;
    a0[e] = l0; a2[e] = l2;
    atomicMax(&segmax[d], encF(fmaxf(l0, l2)));
  }
}

__global__ void attn_z(const float* __restrict__ a0, const float* __restrict__ a2,
                       const int* __restrict__ edst, const unsigned* __restrict__ segmax,
                       float* __restrict__ z, int ne){
  int e = blockIdx.x * blockDim.x + threadIdx.x;
  if (e >= ne) return;
  int d = edst[e];
  float m = decF(segmax[d]);
  atomicAdd(&z[d], expf(a0[e] - m) + expf(a2[e] - m));
}

__global__ void attn_agg(const float* __restrict__ a0, const float* __restrict__ a2,
                         const int* __restrict__ esrc, const int* __restrict__ edst,
                         const unsigned* __restrict__ segmax, const float* __restrict__ z,
                         const float* __restrict__ v0, const float* __restrict__ v2,
                         float* __restrict__ agg, int ne){
  int e = blockIdx.x * (blockDim.x >> 5) + (threadIdx.x >> 5);
  if (e >= ne) return;
  int lane = threadIdx.x & 31;
  int s = esrc[e], d = edst[e];
  float m  = decF(segmax[d]);
  float zz = z[d] + 1e-16f;
  float w0 = expf(a0[e] - m) / zz;
  float w2 = expf(a2[e] - m) / zz;
  const float* pv0 = v0 + (long)s * DIMS;
  const float* pv2 = v2 + (long)e * DIMS;
  float* pa = agg + (long)d * DIMS;
  for (int i = lane; i < DIMS; i += 32)
    atomicAdd(&pa[i], w0 * pv0[i] + w2 * pv2[i]);
}

// h = sigmoid(skip)*out + (1-sigmoid(skip))*emb[ids]   (no restrict: may run in place)
__global__ void skip_mix(const float* outb, const int* __restrict__ ids,
                         const float* __restrict__ emb, const float* __restrict__ skip,
                         float* h, long total){
  long i = (long)blockIdx.x * blockDim.x + threadIdx.x;
  if (i >= total) return;
  float sg = 1.0f / (1.0f + expf(-skip[0]));
  long row = i / DIMS; int d = (int)(i % DIMS);
  h[i] = sg * outb[i] + (1.0f - sg) * emb[(long)ids[row] * DIMS + d];
}

// ---------------- GATv2 root head ----------------
__global__ void xlr_kernel(const float* __restrict__ h_n,
                           const float* __restrict__ Wl, const float* __restrict__ bl,
                           const float* __restrict__ Wr, const float* __restrict__ br,
                           float* __restrict__ xl, float* __restrict__ xr, int n){
  int i = blockIdx.x * blockDim.x + threadIdx.x;
  if (i >= n) return;
  const float* r = h_n + (long)i * DIMS;
  float al = 0.0f, ar = 0.0f;
  for (int d = 0; d < DIMS; ++d){ float v = r[d]; al += v * Wl[d]; ar += v * Wr[d]; }
  xl[i] = al + bl[0]; xr[i] = ar + br[0];
}

__global__ void gat_logits(const float* __restrict__ h_e, const float* __restrict__ We,
                           const float* __restrict__ att,
                           const float* __restrict__ xl, const float* __restrict__ xr,
                           const int* __restrict__ esrc, const int* __restrict__ edst,
                           float* __restrict__ glog, unsigned* __restrict__ gmax, int ne){
  int e = blockIdx.x * blockDim.x + threadIdx.x;
  if (e >= ne) return;
  const float* r = h_e + (long)e * DIMS;
  float hw = 0.0f;
  for (int d = 0; d < DIMS; ++d) hw += r[d] * We[d];
  float mm = xl[esrc[e]] + xr[edst[e]] + hw;
  mm = (mm > 0.0f) ? mm : 0.2f * mm;          // leaky_relu 0.2
  float lg = mm * att[0];
  glog[e] = lg;
  atomicMax(&gmax[edst[e]], encF(lg));
}

__global__ void gat_z(const float* __restrict__ glog, const int* __restrict__ edst,
                      const unsigned* __restrict__ gmax, float* __restrict__ gz, int ne){
  int e = blockIdx.x * blockDim.x + threadIdx.x;
  if (e >= ne) return;
  int d = edst[e];
  atomicAdd(&gz[d], expf(glog[e] - decF(gmax[d])));
}

__global__ void gat_rl(const float* __restrict__ glog, const int* __restrict__ esrc,
                       const int* __restrict__ edst, const unsigned* __restrict__ gmax,
                       const float* __restrict__ gz, const float* __restrict__ xl,
                       float* __restrict__ rl, int ne){
  int e = blockIdx.x * blockDim.x + threadIdx.x;
  if (e >= ne) return;
  int d = edst[e];
  float ga = expf(glog[e] - decF(gmax[d])) / (gz[d] + 1e-16f);
  atomicAdd(&rl[d], ga * xl[esrc[e]]);
}

// ---------------- global reductions over NN (root log_softmax + argmax) ----------------
__global__ void red_max(const float* __restrict__ x, int n, unsigned* __restrict__ gm){
  __shared__ float red[256];
  int tid = threadIdx.x;
  long i = (long)blockIdx.x * 256 + tid;
  red[tid] = (i < n) ? x[i] : -INFINITY;
  __syncthreads();
  for (int s = 128; s; s >>= 1){ if (tid < s) red[tid] = fmaxf(red[tid], red[tid + s]); __syncthreads(); }
  if (tid == 0) atomicMax(gm, encF(red[0]));
}

__global__ void red_sum_arg(const float* __restrict__ x, int n, const unsigned* __restrict__ gm,
                            float* __restrict__ gsum, int* __restrict__ argmin){
  __shared__ float red[256];
  int tid = threadIdx.x;
  long i = (long)blockIdx.x * 256 + tid;
  float mx = decF(*gm);
  float v = (i < n) ? x[i] : -INFINITY;
  if (i < n && v == mx) atomicMin(argmin, (int)i);
  red[tid] = (i < n) ? expf(v - mx) : 0.0f;
  __syncthreads();
  for (int s = 128; s; s >>= 1){ if (tid < s) red[tid] += red[tid + s]; __syncthreads(); }
  if (tid == 0) atomicAdd(gsum, red[0]);
}

__global__ void root_write(const float* __restrict__ x, int n, const unsigned* __restrict__ gm,
                           const float* __restrict__ gsum, float* __restrict__ out){
  long i = (long)blockIdx.x * 256 + threadIdx.x;
  if (i >= n) return;
  out[i] = x[i] - decF(*gm) - logf(*gsum);
}

// ---------------- per-feature softmax aggregation (gn, ge) ----------------
__global__ void feat_max(const float* __restrict__ H, int M, unsigned* __restrict__ fmax){
  int d = threadIdx.x;
  if (d >= DIMS) return;
  long start = (long)blockIdx.x * 2048;
  long end = start + 2048; if (end > M) end = M;
  float m = -INFINITY;
  for (long i = start; i < end; ++i) m = fmaxf(m, H[i * DIMS + d]);
  atomicMax(&fmax[d], encF(m));
}

__global__ void feat_sum(const float* __restrict__ H, int M, const unsigned* __restrict__ fmax,
                         float* __restrict__ s1, float* __restrict__ s2){
  int d = threadIdx.x;
  if (d >= DIMS) return;
  long start = (long)blockIdx.x * 2048;
  long end = start + 2048; if (end > M) end = M;
  float m = decF(fmax[d]);
  float a = 0.0f, b = 0.0f;
  for (long i = start; i < end; ++i){
    float x = H[i * DIMS + d];
    float e_ = expf(x - m);
    a += e_; b += e_ * x;
  }
  atomicAdd(&s1[d], a); atomicAdd(&s2[d], b);
}

__global__ void feat_final(const float* __restrict__ s1, const float* __restrict__ s2,
                           float* __restrict__ g){
  int d = threadIdx.x;
  if (d < 2 * DIMS) g[d] = s2[d] / s1[d];
}

// ---------------- frame prediction (single block) ----------------
__global__ void frame_kernel(const float* __restrict__ g, const float* __restrict__ Wf,
                             const float* __restrict__ bf, float* __restrict__ out){
  __shared__ float t[NFR];
  __shared__ float red[256];
  int tid = threadIdx.x;
  for (int f = tid; f < NFR; f += 256){
    float acc = bf[f];
    for (int d = 0; d < 2 * DIMS; ++d) acc += g[d] * Wf[(long)d * NFR + f];
    t[f] = acc;
  }
  __syncthreads();
  float m = -INFINITY;
  for (int f = tid; f < NFR; f += 256) m = fmaxf(m, t[f]);
  red[tid] = m; __syncthreads();
  for (int s = 128; s; s >>= 1){ if (tid < s) red[tid] = fmaxf(red[tid], red[tid + s]); __syncthreads(); }
  m = red[0]; __syncthreads();
  float sum = 0.0f;
  for (int f = tid; f < NFR; f += 256) sum += expf(t[f] - m);
  red[tid] = sum; __syncthreads();
  for (int s = 128; s; s >>= 1){ if (tid < s) red[tid] += red[tid + s]; __syncthreads(); }
  float lse = m + logf(red[0]);
  for (int f = tid; f < NFR; f += 256) out[f] = t[f] - lse;
}

// role bias: cvec[r] = brole[r] + sum_d gn[d] * Wrole[100+d, r]
__global__ void cvec_kernel(const float* __restrict__ g, const float* __restrict__ Wrole,
                            const float* __restrict__ brole, float* __restrict__ cvec){
  int r = threadIdx.x;
  if (r >= NRL) return;
  float acc = brole[r];
  for (int d = 0; d < DIMS; ++d) acc += g[d] * Wrole[(long)(DIMS + d) * NRL + r];
  cvec[r] = acc;
}

// role mask + row log_softmax over 32 (one wave per edge, lane == role)
__global__ void role_kernel(float* role, const int* __restrict__ esrc,
                            const int* __restrict__ rootIdx, int ne){
  int e = blockIdx.x * (blockDim.x >> 5) + (threadIdx.x >> 5);
  if (e >= ne) return;
  int lane = threadIdx.x & 31;
  float v = (esrc[e] == *rootIdx) ? role[(long)e * NRL + lane] : 0.0f;
  float m = v;
  for (int o = 16; o; o >>= 1) m = fmaxf(m, __shfl_xor(m, o, 32));
  float ex = expf(v - m), s = ex;
  for (int o = 16; o; o >>= 1) s += __shfl_xor(s, o, 32);
  role[(long)e * NRL + lane] = v - m - logf(s);
}

// =====================================================================
extern "C" void kernel_launch(void* const* d_in, const int* in_sizes, int n_in,
                              void* d_out, int out_size, void* d_ws, size_t ws_size,
                              hipStream_t stream){
  const int NN = in_sizes[0];
  const int NE = in_sizes[1];

  const int*   x_node = (const int*)  d_in[0];
  const int*   x_edge = (const int*)  d_in[1];
  const int*   esrc   = (const int*)  d_in[2];
  const int*   edst   = (const int*)  d_in[3];
  const float* emb    = (const float*)d_in[4];
  const float* Wkqv_n = (const float*)d_in[5];  const float* bkqv_n = (const float*)d_in[6];
  const float* Wkqv_e = (const float*)d_in[7];  const float* bkqv_e = (const float*)d_in[8];
  const float* a_k    = (const float*)d_in[9];  const float* a_v    = (const float*)d_in[10];
  const float* p_rel  = (const float*)d_in[11];
  const float* Wout_n = (const float*)d_in[12]; const float* bout_n = (const float*)d_in[13];
  const float* Wout_e = (const float*)d_in[14]; const float* bout_e = (const float*)d_in[15];
  const float* skip_n = (const float*)d_in[16]; const float* skip_e = (const float*)d_in[17];
  const float* Wl     = (const float*)d_in[18]; const float* bl     = (const float*)d_in[19];
  const float* Wr     = (const float*)d_in[20]; const float* br     = (const float*)d_in[21];
  const float* We     = (const float*)d_in[22]; const float* att    = (const float*)d_in[23];
  // d_in[24] = b_gat: constant shift over all root logits -> cancels in log_softmax & argmax
  const float* Wf     = (const float*)d_in[25]; const float* bf     = (const float*)d_in[26];
  const float* Wrole  = (const float*)d_in[27]; const float* brole  = (const float*)d_in[28];

  float* out_root  = (float*)d_out;
  float* out_frame = out_root + NN;
  float* out_role  = out_frame + NFR;

  // ---- carve workspace ----
  char* ws = (char*)d_ws;
  size_t off = 0;
  auto carve = [&](size_t bytes) -> void* {
    void* p = ws + off;
    off = (off + bytes + 255) & ~(size_t)255;
    return p;
  };
  float*    kqv_n  = (float*)   carve((size_t)NN * LDKQV * 4);
  float*    kqv_e  = (float*)   carve((size_t)NE * LDKQV * 4);
  float*    k0     = (float*)   carve((size_t)NN * DIMS * 4);
  float*    v0     = (float*)   carve((size_t)NN * DIMS * 4);
  float*    v1     = (float*)   carve((size_t)NN * DIMS * 4);
  float*    k2     = (float*)   carve((size_t)NE * DIMS * 4);
  float*    v2     = (float*)   carve((size_t)NE * DIMS * 4);
  float*    a0     = (float*)   carve((size_t)NE * 4);
  float*    a2     = (float*)   carve((size_t)NE * 4);
  unsigned* segmax = (unsigned*)carve((size_t)NN * 4);
  float*    z_n    = (float*)   carve((size_t)NN * 4);
  float*    agg_n  = (float*)   carve((size_t)NN * DIMS * 4);
  float*    out_n  = (float*)   carve((size_t)NN * DIMS * 4);
  float*    h_n    = (float*)   carve((size_t)NN * DIMS * 4);
  float*    h_e    = (float*)   carve((size_t)NE * DIMS * 4);   // holds out_e, then h_e
  float*    xl     = (float*)   carve((size_t)NN * 4);
  float*    xr     = (float*)   carve((size_t)NN * 4);
  float*    glog   = (float*)   carve((size_t)NE * 4);
  unsigned* gmax   = (unsigned*)carve((size_t)NN * 4);
  float*    gz     = (float*)   carve((size_t)NN * 4);
  float*    rl     = (float*)   carve((size_t)NN * 4);
  unsigned* fmax   = (unsigned*)carve(200 * 4);
  float*    fs1    = (float*)   carve(200 * 4);
  float*    fs2    = (float*)   carve(200 * 4);
  float*    g      = (float*)   carve(200 * 4);
  float*    cvec   = (float*)   carve(NRL * 4);
  unsigned* rmax   = (unsigned*)carve(4);
  float*    rsum   = (float*)   carve(4);
  int*      ridx   = (int*)     carve(4);

  auto fill = [&](void* p, unsigned v, int n){
    fill_u32<<<(n + 255) / 256, 256, 0, stream>>>((unsigned*)p, v, n);
  };

  // ---- per-call (re)initialization of all accumulators ----
  fill(segmax, NEG_INF_ENC, NN);
  fill(z_n,    0u,          NN);
  fill(agg_n,  0u,          NN * DIMS);
  fill(gmax,   NEG_INF_ENC, NN);
  fill(gz,     0u,          NN);
  fill(rl,     0u,          NN);
  fill(fmax,   NEG_INF_ENC, 200);
  fill(fs1,    0u,          200);
  fill(fs2,    0u,          200);
  fill(rmax,   NEG_INF_ENC, 1);
  fill(rsum,   0u,          1);
  fill(ridx,   0x7FFFFFFFu, 1);

  dim3 blk(128);                                  // 4 waves / block
  const int smt_n = (NN + 31) / 32;               // 32-row supertiles (MT=2)
  const int smt_e = (NE + 31) / 32;
  dim3 gN300((smt_n + 3) / 4, 5);                 // N=300, NT=4 -> 5 col groups
  dim3 gE300((smt_e + 3) / 4, 5);
  dim3 gN100((smt_n + 3) / 4, 2);                 // N=100, NT=4 -> 2 col groups
  dim3 gE100((smt_e + 3) / 4, 2);
  dim3 gE32 ((smt_e + 3) / 4, 1);                 // N=32,  NT=2 -> 1 col group

  // ---- kqv projections (gather emb rows) ----
  gemm_k100_wmma<2,4,LDKQV><<<gN300, blk, 0, stream>>>(
      emb, DIMS, x_node, Wkqv_n, bkqv_n, kqv_n, LDKQV, NN, LDKQV);
  gemm_k100_wmma<2,4,LDKQV><<<gE300, blk, 0, stream>>>(
      emb, DIMS, x_edge, Wkqv_e, bkqv_e, kqv_e, LDKQV, NE, LDKQV);

  // ---- relation transforms (k1 provably unnecessary; q_e unused) ----
  gemm_k100_wmma<2,4,DIMS><<<gN100, blk, 0, stream>>>(
      kqv_n,       LDKQV, (const int*)nullptr, a_k,         (const float*)nullptr, k0, DIMS, NN, DIMS);
  gemm_k100_wmma<2,4,DIMS><<<gN100, blk, 0, stream>>>(
      kqv_n + 200, LDKQV, (const int*)nullptr, a_v,         (const float*)nullptr, v0, DIMS, NN, DIMS);
  gemm_k100_wmma<2,4,DIMS><<<gN100, blk, 0, stream>>>(
      kqv_n + 200, LDKQV, (const int*)nullptr, a_v + 10000, (const float*)nullptr, v1, DIMS, NN, DIMS);
  gemm_k100_wmma<2,4,DIMS><<<gE100, blk, 0, stream>>>(
      kqv_e,       LDKQV, (const int*)nullptr, a_k + 20000, (const float*)nullptr, k2, DIMS, NE, DIMS);
  gemm_k100_wmma<2,4,DIMS><<<gE100, blk, 0, stream>>>(
      kqv_e + 200, LDKQV, (const int*)nullptr, a_v + 20000, (const float*)nullptr, v2, DIMS, NE, DIMS);

  // ---- HGT segment softmax over node destinations + aggregation ----
  attn_logits<<<(NE + 3) / 4, blk, 0, stream>>>(kqv_n, k0, k2, esrc, edst, p_rel, a0, a2, segmax, NE);
  attn_z     <<<(NE + 255) / 256, 256, 0, stream>>>(a0, a2, edst, segmax, z_n, NE);
  attn_agg   <<<(NE + 3) / 4, blk, 0, stream>>>(a0, a2, esrc, edst, segmax, z_n, v0, v2, agg_n, NE);

  // ---- GELU hoisted out of GEMMs: in-place on agg_n; and on v1 (gelu∘gather == gather∘gelu,
  //      and the type-1 softmax weight 1/(1+1e-16) rounds to exactly 1.0f) ----
  {
    long tn = (long)NN * DIMS;
    gelu_inplace<<<(unsigned)((tn + 255) / 256), 256, 0, stream>>>(agg_n, tn);
    gelu_inplace<<<(unsigned)((tn + 255) / 256), 256, 0, stream>>>(v1, tn);
  }

  // ---- output projections (pure WMMA GEMMs); out_e gathers gelu(v1)[src] ----
  gemm_k100_wmma<2,4,DIMS><<<gN100, blk, 0, stream>>>(
      agg_n, DIMS, (const int*)nullptr, Wout_n, bout_n, out_n, DIMS, NN, DIMS);
  gemm_k100_wmma<2,4,DIMS><<<gE100, blk, 0, stream>>>(
      v1, DIMS, esrc, Wout_e, bout_e, h_e, DIMS, NE, DIMS);

  // ---- skip mixing ----
  {
    long tn = (long)NN * DIMS, te = (long)NE * DIMS;
    skip_mix<<<(unsigned)((tn + 255) / 256), 256, 0, stream>>>(out_n, x_node, emb, skip_n, h_n, tn);
    skip_mix<<<(unsigned)((te + 255) / 256), 256, 0, stream>>>(h_e,  x_edge, emb, skip_e, h_e, te);
  }

  // ---- GATv2 root head ----
  xlr_kernel <<<(NN + 255) / 256, 256, 0, stream>>>(h_n, Wl, bl, Wr, br, xl, xr, NN);
  gat_logits <<<(NE + 255) / 256, 256, 0, stream>>>(h_e, We, att, xl, xr, esrc, edst, glog, gmax, NE);
  gat_z      <<<(NE + 255) / 256, 256, 0, stream>>>(glog, edst, gmax, gz, NE);
  gat_rl     <<<(NE + 255) / 256, 256, 0, stream>>>(glog, esrc, edst, gmax, gz, xl, rl, NE);

  red_max    <<<(NN + 255) / 256, 256, 0, stream>>>(rl, NN, rmax);
  red_sum_arg<<<(NN + 255) / 256, 256, 0, stream>>>(rl, NN, rmax, rsum, ridx);
  root_write <<<(NN + 255) / 256, 256, 0, stream>>>(rl, NN, rmax, rsum, out_root);

  // ---- per-feature softmax aggregation -> gn, ge ----
  feat_max <<<(NN + 2047) / 2048, 128, 0, stream>>>(h_n, NN, fmax);
  feat_max <<<(NE + 2047) / 2048, 128, 0, stream>>>(h_e, NE, fmax + DIMS);
  feat_sum <<<(NN + 2047) / 2048, 128, 0, stream>>>(h_n, NN, fmax, fs1, fs2);
  feat_sum <<<(NE + 2047) / 2048, 128, 0, stream>>>(h_e, NE, fmax + DIMS, fs1 + DIMS, fs2 + DIMS);
  feat_final<<<1, 256, 0, stream>>>(fs1, fs2, g);

  // ---- frame prediction ----
  frame_kernel<<<1, 256, 0, stream>>>(g, Wf, bf, out_frame);

  // ---- role head: GEMM with bias folding in the gn-half + brole ----
  cvec_kernel<<<1, 32, 0, stream>>>(g, Wrole, brole, cvec);
  gemm_k100_wmma<2,2,NRL><<<gE32, blk, 0, stream>>>(
      h_e, DIMS, (const int*)nullptr, Wrole, cvec, out_role, NRL, NE, NRL);
  role_kernel<<<(NE + 3) / 4, blk, 0, stream>>>(out_role, esrc, ridx, NE);
}